// LSTM_40948218200110
// MI455X (gfx1250) — compile-verified
//
#include <hip/hip_runtime.h>

// CDNA5 / gfx1250 fused LSTM (software-pipelined):
//   gates^T[16 gate x 16 batch] = W[16x12] * [x;h][12x16] via
//   V_WMMA_F32_16X16X4_F32 (exact fp32, accumulation order bias->x0->x1->h).
//   The two x-WMMAs for step t+1 are computed in the shadow of step t's cell
//   update, so the serial recurrence chain per step is only:
//   wmma(W_hh*h) -> v_tanh (act) -> v_permlanex16 -> fma(c) -> v_tanh(c) -> h.
//   Both lane-halves redundantly carry c/h so no back-swaps are needed.
// One wave (32 lanes) owns a 16-batch tile for the full T-step recurrence.

typedef __attribute__((ext_vector_type(2))) float v2f;
typedef __attribute__((ext_vector_type(8))) float v8f;

#define SWZ_SWAPX16 0x401F   // ds_swizzle fallback: and=0x1f,or=0,xor=0x10

__device__ __forceinline__ float tanh_t(float x) {
#if __has_builtin(__builtin_amdgcn_tanhf)
  return __builtin_amdgcn_tanhf(x);            // v_tanh_f32 (TRANS)
#else
  float r;
  asm volatile("v_tanh_f32 %0, %1\n\tv_nop" : "=v"(r) : "v"(x));
  return r;
#endif
}

// lane <- lane ^ 16 (swap the two 16-lane halves of the wave32)
__device__ __forceinline__ float swapx16(float x) {
  int i = __builtin_bit_cast(int, x);
#if __has_builtin(__builtin_amdgcn_permlanex16)
  int r = __builtin_amdgcn_permlanex16(i, i, 0x76543210, 0xfedcba98, true, false);
#else
  int r = __builtin_amdgcn_ds_swizzle(i, SWZ_SWAPX16);
#endif
  return __builtin_bit_cast(float, r);
}

__device__ __forceinline__ v8f wmma4(v2f A, v2f B, v8f C) {
  return __builtin_amdgcn_wmma_f32_16x16x4_f32(false, A, false, B, (short)0, C,
                                               false, false);
}

__global__ void __launch_bounds__(32)
lstm_wmma_fused(const float* __restrict__ x,
                const float* __restrict__ W_ih,
                const float* __restrict__ W_hh,
                const float* __restrict__ b_ih,
                const float* __restrict__ b_hh,
                const float* __restrict__ fc_w,
                const float* __restrict__ fc_b,
                float* __restrict__ out,
                int T) {
  const int  lane = threadIdx.x & 31;
  const int  tile = blockIdx.x;        // 16-batch tile
  const int  n    = lane & 15;         // batch column within tile
  const bool lo   = lane < 16;
  const int  b    = tile * 16 + n;

  // ---- A-layout weights (M = gate = lane&15; VGPR v + lane-half -> K) ----
  // A 16x4 f32: vgpr0 = K0 (lanes 0-15) / K2 (lanes 16-31); vgpr1 = K1 / K3.
  const int m   = n;                   // gate row 0..15 (PyTorch order i,f,g,o)
  const int k01 = lo ? 0 : 2;
  v2f A0, A1, A2;
  A0.x = W_ih[m * 8 + (k01 + 0)];      // x gates, K = 0..3
  A0.y = W_ih[m * 8 + (k01 + 1)];
  A1.x = W_ih[m * 8 + 4 + (k01 + 0)];  // x gates, K = 4..7
  A1.y = W_ih[m * 8 + 4 + (k01 + 1)];
  A2.x = W_hh[m * 4 + (k01 + 0)];      // h gates, K = 8..11
  A2.y = W_hh[m * 4 + (k01 + 1)];

  // ---- bias in C/D layout: vgpr r -> gate r (lower half) / r+8 (upper) ----
  v8f bias;
  {
    const int gb = lo ? 0 : 8;
#pragma unroll
    for (int r = 0; r < 8; ++r) bias[r] = b_ih[gb + r] + b_hh[gb + r];
  }

  // Lane-selected activation constants for D vgprs 0..3:
  //   lower half = i gate -> sigmoid(x) = 0.5 + 0.5*tanh(0.5x)
  //   upper half = g gate -> tanh(x)
  const float s_pre = lo ? 0.5f : 1.0f;
  const float s_mul = lo ? 0.5f : 1.0f;
  const float s_add = lo ? 0.5f : 0.0f;

  // Per-lane x stream (B-layout slice).
  // B 4x16 f32: vgpr0 = K0 (lanes 0-15) / K2 (16-31); vgpr1 = K1 / K3.
  const float* xp = x + (size_t)b * (size_t)T * 8 + k01;

  v2f Bh; Bh.x = 0.0f; Bh.y = 0.0f;                    // h in B layout
  float c0 = 0.f, c1 = 0.f, c2 = 0.f, c3 = 0.f;        // cell, valid on BOTH halves

  // Prologue: x contribution for t = 0.
  v8f gx;
  {
    v2f Bx0 = *(const v2f*)(xp);
    v2f Bx1 = *(const v2f*)(xp + 4);
    gx = wmma4(A0, Bx0, bias);
    gx = wmma4(A1, Bx1, gx);
  }

  for (int t = 0; t < T; ++t) {
    // Issue next step's loads early (clamped, branchless at t = T-1).
    int tn = t + 1; if (tn >= T) tn = T - 1;
    const float* xn = xp + (size_t)tn * 8;
    v2f Nx0 = *(const v2f*)(xn);
    v2f Nx1 = *(const v2f*)(xn + 4);
    int tp = t + 8; if (tp >= T) tp = T - 1;
    __builtin_prefetch(xp + (size_t)tp * 8, 0, 3);

    // Critical-path WMMA: gates = gx + W_hh * h
    v8f g = wmma4(A2, Bh, gx);

    // Activations: vgpr 0..3 -> sigmoid(i) [lower] / tanh(g) [upper]
    //              vgpr 4..7 -> sigmoid(f) [lower] / sigmoid(o) [upper]
    float ai[4], af[4];
#pragma unroll
    for (int j = 0; j < 4; ++j) ai[j] = fmaf(s_mul, tanh_t(s_pre * g[j]), s_add);
#pragma unroll
    for (int j = 0; j < 4; ++j) af[j] = fmaf(0.5f, tanh_t(0.5f * g[4 + j]), 0.5f);

    // Off-critical-path: x contribution for step t+1 overlaps the cell math.
    v8f gxn = wmma4(A0, Nx0, bias);
    gxn = wmma4(A1, Nx1, gxn);

    // Single exchange stage: both halves get the other half's activations.
#pragma unroll
    for (int j = 0; j < 4; ++j) {
      const float ais = swapx16(ai[j]);     // lower: tanh(g) | upper: sig(i)
      const float afs = swapx16(af[j]);     // lower: sig(o)  | upper: sig(f)
      const float p   = ai[j] * ais;        // sig(i)*tanh(g) on BOTH halves
      const float fj  = lo ? af[j] : afs;   // sigmoid(f)
      const float oj  = lo ? afs : af[j];   // sigmoid(o)
      float& cj = (j == 0) ? c0 : (j == 1) ? c1 : (j == 2) ? c2 : c3;
      cj = fmaf(fj, cj, p);                 // c = f*c + i*g   (both halves)
      const float hj = oj * tanh_t(cj);     // h = o*tanh(c)   (both halves)
      // Assemble next-step B layout locally (no back-swaps):
      //   Bh.x: lower = h[b][0], upper = h[b][2]
      //   Bh.y: lower = h[b][1], upper = h[b][3]
      if (j == 0) { if (lo) Bh.x = hj; }
      else if (j == 1) { if (lo) Bh.y = hj; }
      else if (j == 2) { if (!lo) Bh.x = hj; }
      else             { if (!lo) Bh.y = hj; }
    }

    gx = gxn;
  }

  // Final projection: out[b] = sum_j h[b][j]*fc_w[j] + fc_b
  const float wA = lo ? fc_w[0] : fc_w[2];
  const float wB = lo ? fc_w[1] : fc_w[3];
  float p  = Bh.x * wA + Bh.y * wB;    // lower: h0*w0+h1*w1, upper: h2*w2+h3*w3
  float ps = swapx16(p);
  if (lo) out[b] = p + ps + fc_b[0];
}

extern "C" void kernel_launch(void* const* d_in, const int* in_sizes, int n_in,
                              void* d_out, int out_size, void* d_ws, size_t ws_size,
                              hipStream_t stream) {
  (void)n_in; (void)d_ws; (void)ws_size;
  const float* x    = (const float*)d_in[0];
  const float* W_ih = (const float*)d_in[1];
  const float* W_hh = (const float*)d_in[2];
  const float* b_ih = (const float*)d_in[3];
  const float* b_hh = (const float*)d_in[4];
  const float* fc_w = (const float*)d_in[5];
  const float* fc_b = (const float*)d_in[6];
  float* out = (float*)d_out;

  const int B = out_size;                       // 4096
  const int T = in_sizes[0] / (B * 8);          // 1024
  const int tiles = B / 16;                     // 256 waves, one per 16-batch tile

  lstm_wmma_fused<<<tiles, 32, 0, stream>>>(x, W_ih, W_hh, b_ih, b_hh,
                                            fc_w, fc_b, out, T);
}